// HyperbolicLinearHyperboloid_18743237279913
// MI455X (gfx1250) — compile-verified
//
#include <hip/hip_runtime.h>
#include <math.h>

typedef __attribute__((ext_vector_type(2))) float v2f;
typedef __attribute__((ext_vector_type(4))) float v4f;
typedef __attribute__((ext_vector_type(8))) float v8f;

#define SP 64
#define DIM 65
#define ROWS_PER_BLOCK 128
#define XPAD 68      // row stride (floats) for staged sp -> conflict-free A-frag b64 loads
#define WSTRIDE 160  // k-pair stride (floats) for packed weight; 160%64==32 -> halves hit disjoint banks

__global__ __launch_bounds__(128)
void hyp_linear_kernel(const float* __restrict__ x,
                       const float* __restrict__ w,
                       const float* __restrict__ bias,
                       float* __restrict__ out,
                       int nrows)
{
    __shared__ __align__(16) float xs[ROWS_PER_BLOCK * XPAD];   // sp values, padded rows
    __shared__ __align__(16) float wp[32 * WSTRIDE];            // weight, B-fragment pair-packed
    __shared__ __align__(16) float xt[ROWS_PER_BLOCK];          // time coords
    __shared__ __align__(16) float coef[ROWS_PER_BLOCK];        // arccosh(x0)/||sp||

    const int t    = threadIdx.x;
    const int wave = t >> 5;
    const int lane = t & 31;
    const int r16  = lane & 15;
    const int hh   = lane >> 4;

    // ---- stage weight: wp[(i>>1)*WSTRIDE + o*2 + (i&1)] = w[o*64+i]  (B[k][n] pair-packed)
    for (int j = 0; j < 32; ++j) {
        int idx = t + j * 128;
        int o = idx >> 6, i = idx & 63;
        wp[(i >> 1) * WSTRIDE + o * 2 + (i & 1)] = w[idx];
    }

    // ---- stage x rows (coalesced global -> LDS), split time / space
    long gbase = (long)blockIdx.x * ROWS_PER_BLOCK;
    long ebase = gbase * DIM;
    long etot  = (long)nrows * DIM;
    for (int j = 0; j < 65; ++j) {
        int idx = t + j * 128;
        long ge = ebase + idx;
        float v = (ge < etot) ? x[ge] : 0.0f;
        int row = idx / 65;
        int col = idx - row * 65;
        if (col == 0) xt[row] = v;
        else          xs[row * XPAD + (col - 1)] = v;
    }
    __syncthreads();

    // ---- per-row logmap coefficient (one row per thread)
    {
        float ss = 0.f;
        const v4f* rp4 = (const v4f*)&xs[t * XPAD];
        #pragma unroll
        for (int i = 0; i < SP / 4; ++i) {
            v4f q = rp4[i];
            ss = fmaf(q.x, q.x, ss);
            ss = fmaf(q.y, q.y, ss);
            ss = fmaf(q.z, q.z, ss);
            ss = fmaf(q.w, q.w, ss);
        }
        float n  = fmaxf(sqrtf(ss), 1e-15f);
        float th = fmaxf(xt[t], 1.0f + 1e-7f);
        coef[t] = acoshf(th) / n;
    }
    __syncthreads();

    // ---- bias fragments and Sum(bias^2) (uniform across rows)
    float bn[4];
    #pragma unroll
    for (int nt = 0; nt < 4; ++nt) bn[nt] = bias[nt * 16 + r16];
    float bsq = 0.f;
    #pragma unroll
    for (int nt = 0; nt < 4; ++nt) bsq = fmaf(bn[nt], bn[nt], bsq);
    #pragma unroll
    for (int m = 1; m <= 8; m <<= 1) bsq += __shfl_xor(bsq, m, 32);

    // ---- two 16-row M-tiles per wave
    for (int mt = 0; mt < 2; ++mt) {
        const int mb = wave * 32 + mt * 16;
        v8f acc[4] = {};

        // A frag: lane holds xs[row=mb+r16][k0 + 2*hh + {0,1}]  (ISA 16x4 f32 A layout)
        const float* ap = &xs[(mb + r16) * XPAD + 2 * hh];
        #pragma unroll
        for (int k4 = 0; k4 < 16; ++k4) {
            v2f a = *(const v2f*)(ap + 4 * k4);
            #pragma unroll
            for (int nt = 0; nt < 4; ++nt) {
                v2f b = *(const v2f*)(&wp[(2 * k4 + hh) * WSTRIDE + (nt * 16 + r16) * 2]);
                acc[nt] = __builtin_amdgcn_wmma_f32_16x16x4_f32(
                    false, a, false, b, (short)0, acc[nt], false, false);
            }
        }

        // scale by per-row coef: D layout row = mb + 8*hh + v
        v4f c0 = *(const v4f*)(&coef[mb + 8 * hh]);
        v4f c1 = *(const v4f*)(&coef[mb + 8 * hh + 4]);
        float cz[8] = { c0.x, c0.y, c0.z, c0.w, c1.x, c1.y, c1.z, c1.w };

        float psq[8], pdb[8];
        #pragma unroll
        for (int v = 0; v < 8; ++v) { psq[v] = 0.f; pdb[v] = 0.f; }
        #pragma unroll
        for (int nt = 0; nt < 4; ++nt) {
            #pragma unroll
            for (int v = 0; v < 8; ++v) {
                float hv = acc[nt][v] * cz[v];
                acc[nt][v] = hv;
                psq[v] = fmaf(hv, hv, psq[v]);
                pdb[v] = fmaf(hv, bn[nt], pdb[v]);
            }
        }
        // reduce over the 16 lanes holding each row (xor masks stay inside half-wave)
        #pragma unroll
        for (int m = 1; m <= 8; m <<= 1) {
            #pragma unroll
            for (int v = 0; v < 8; ++v) {
                psq[v] += __shfl_xor(psq[v], m, 32);
                pdb[v] += __shfl_xor(pdb[v], m, 32);
            }
        }

        // ---- lane-parallel scalar epilogue: lane computes ONE row's chain (v = r16&7),
        //      instead of every lane redundantly computing all 8 rows.
        float Af, Bf, ot;
        {
            const int vs = r16 & 7;
            float ssq = psq[vs], dhb = pdb[vs];
            // expmap_0
            float n1  = fmaxf(sqrtf(ssq), 1e-15f);
            float th1 = fminf(n1, 15.0f);
            float s1  = sinhf(th1) / n1;          // m_sp = s1 * h
            float msq = s1 * s1 * ssq;
            float mtv = sqrtf(1.0f + msq);        // m_t after proj
            // ptransp_0 + proj_tan
            float yn    = fmaxf(sqrtf(msq), 1e-15f);
            float syn   = s1 / yn;
            float alpha = syn * dhb;
            float k2    = alpha * (1.0f - mtv) * syn;   // res_sp = bias - k2*h
            float tt    = s1 * (dhb - k2 * ssq) / fmaxf(mtv, 1e-15f);
            // expmap on tangent vector [tt, bias - k2*h]
            float rsq = bsq - 2.0f * k2 * dhb + k2 * k2 * ssq;  // Sum(res_sp^2)
            float md  = rsq - tt * tt;                          // Minkowski dot
            float un  = sqrtf(fmaxf(md, 1e-15f));
            float th2 = fminf(fmaxf(un, 1e-15f), 15.0f);
            float ch2 = coshf(th2);
            float s2t = sinhf(th2) / th2;
            Af = ch2 * s1 - s2t * k2;              // out_sp = Af*h + Bf*bias
            Bf = s2t;
            float osq = Af * Af * ssq + 2.f * Af * Bf * dhb + Bf * Bf * bsq;
            ot = sqrtf(1.0f + osq);                // final proj time coord
        }

        // broadcast each row's (Af, Bf, ot) from its owner lane within this half-wave
        const int srcbase = hh * 16;
        const long growb = gbase + mb + 8 * hh;
        #pragma unroll
        for (int v = 0; v < 8; ++v) {
            float Afv = __shfl(Af, srcbase + v, 32);
            float Bfv = __shfl(Bf, srcbase + v, 32);
            float otv = __shfl(ot, srcbase + v, 32);
            long row = growb + v;
            if (row < nrows) {
                float* orow = out + row * DIM;
                if (r16 == 0) orow[0] = otv;
                #pragma unroll
                for (int nt = 0; nt < 4; ++nt)
                    orow[1 + nt * 16 + r16] = fmaf(Afv, acc[nt][v], Bfv * bn[nt]);
            }
        }
    }
}

extern "C" void kernel_launch(void* const* d_in, const int* in_sizes, int n_in,
                              void* d_out, int out_size, void* d_ws, size_t ws_size,
                              hipStream_t stream) {
    const float* x    = (const float*)d_in[0];
    const float* w    = (const float*)d_in[1];
    const float* bias = (const float*)d_in[2];
    float* out = (float*)d_out;
    int nrows = in_sizes[0] / DIM;
    int grid  = (nrows + ROWS_PER_BLOCK - 1) / ROWS_PER_BLOCK;
    hipLaunchKernelGGL(hyp_linear_kernel, dim3(grid), dim3(128), 0, stream,
                       x, w, bias, out, nrows);
}